// MPConv2d_53506702573941
// MI455X (gfx1250) — compile-verified
//
#include <hip/hip_runtime.h>

// MPConv2d on gfx1250: sparsemax-threshold conv via exact bisection.
// v(h,o) = patch[h] +/- W[h][o] tiles are formed with v_wmma_f32_16x16x4_f32
// (A = patch column, B = ones row, C = W tile), threshold sums on VALU.

typedef __attribute__((ext_vector_type(2))) float v2f;
typedef __attribute__((ext_vector_type(8))) float v8f;

#define HP     288           // C_IN * 3 * 3
#define WPAD   34            // LDS row pitch for W^T (rows 8 apart -> disjoint banks)
#define NITER  26            // bisection iterations (2^-26 of bracket ~ fp32 exact)
#define GAMMA  1.0f
#define DTOT   576.0f        // 2 * HP

__global__ __launch_bounds__(256) void mpconv_sparsemax_kernel(
    const float* __restrict__ x,      // [2,32,32,32]
    const float* __restrict__ w,      // [32,32,3,3] -> flat [o][h]
    float* __restrict__ out)          // [2,32,32,32]
{
    __shared__ float sW[HP * WPAD];   // W^T: sW[h*WPAD + o]
    __shared__ float sP[8][HP];       // per-wave patch

    const int tid  = threadIdx.x;
    const int lane = tid & 31;
    const int wv   = tid >> 5;

    // ---- stage weights transposed (global [o*288+h] -> LDS [h][o]) ----
    for (int g = tid; g < 32 * HP; g += 256) {
        int o = g / HP;
        int h = g - o * HP;
        sW[h * WPAD + o] = w[g];
    }

    // ---- stage this wave's patch (lane = input channel c) ----
    const int loc = blockIdx.x * 8 + wv;         // 0..2047
    const int b   = loc >> 10;
    const int pos = loc & 1023;
    const int oh  = pos >> 5, ow = pos & 31;
    {
        const int c = lane;
        const float* xb = x + ((b * 32 + c) * 32) * 32;
        #pragma unroll
        for (int ki = 0; ki < 3; ++ki) {
            #pragma unroll
            for (int kj = 0; kj < 3; ++kj) {
                const int y  = oh - 1 + ki;
                const int xx = ow - 1 + kj;
                float v = 0.0f;
                if ((unsigned)y < 32u && (unsigned)xx < 32u) v = xb[y * 32 + xx];
                sP[wv][c * 9 + ki * 3 + kj] = v;   // h = c*9 + i*3 + j
            }
        }
    }
    __syncthreads();

    const int   half   = lane >> 4;          // which M-half of WMMA tiles this lane holds
    const int   l16    = lane & 15;
    const int   rowoff = half * 8;
    const float bsel   = half ? 0.0f : 1.0f; // B row K=0 == ones (lanes 0-15 of vgpr0)
    v2f Bm; Bm[0] = bsel; Bm[1] = 0.0f;
    const float* pP = sP[wv];

    // ---- pass 1: hi = max_h |patch +/- W| per (side, o-half) ----
    float hi_pa = 0.0f, hi_qa = 0.0f, hi_pb = 0.0f, hi_qb = 0.0f;
    for (int h0 = 0; h0 < HP; h0 += 16) {
        float pv = pP[h0 + l16];
        float a0 = half ? 0.0f : pv;         // A: K=0 column = patch, rest 0
        v2f Ap; Ap[0] =  a0; Ap[1] = 0.0f;
        v2f Aq; Aq[0] = -a0; Aq[1] = 0.0f;   // gives W - patch; |.| == |patch - W|
        v8f Ca, Cb;
        #pragma unroll
        for (int r = 0; r < 8; ++r) {
            const float* row = &sW[(h0 + r + rowoff) * WPAD];
            Ca[r] = row[l16];
            Cb[r] = row[l16 + 16];
        }
        v8f Dpa = __builtin_amdgcn_wmma_f32_16x16x4_f32(false, Ap, false, Bm, (short)0, Ca, false, false);
        v8f Dqa = __builtin_amdgcn_wmma_f32_16x16x4_f32(false, Aq, false, Bm, (short)0, Ca, false, false);
        v8f Dpb = __builtin_amdgcn_wmma_f32_16x16x4_f32(false, Ap, false, Bm, (short)0, Cb, false, false);
        v8f Dqb = __builtin_amdgcn_wmma_f32_16x16x4_f32(false, Aq, false, Bm, (short)0, Cb, false, false);
        #pragma unroll
        for (int r = 0; r < 8; ++r) {
            hi_pa = fmaxf(hi_pa, __builtin_fabsf(Dpa[r]));
            hi_qa = fmaxf(hi_qa, __builtin_fabsf(Dqa[r]));
            hi_pb = fmaxf(hi_pb, __builtin_fabsf(Dpb[r]));
            hi_qb = fmaxf(hi_qb, __builtin_fabsf(Dqb[r]));
        }
    }
    hi_pa = fmaxf(hi_pa, __shfl_xor(hi_pa, 16, 32));
    hi_qa = fmaxf(hi_qa, __shfl_xor(hi_qa, 16, 32));
    hi_pb = fmaxf(hi_pb, __shfl_xor(hi_pb, 16, 32));
    hi_qb = fmaxf(hi_qb, __shfl_xor(hi_qb, 16, 32));

    // ---- bisection on f(t) = sum_h max(|v_h| - t, -2t, 0) = GAMMA ----
    const float lo0 = -GAMMA / DTOT;         // f(lo0) >= DTOT*(GAMMA/DTOT) = GAMMA
    float lo_pa = lo0, lo_qa = lo0, lo_pb = lo0, lo_qb = lo0;

    for (int it = 0; it < NITER; ++it) {
        const float t_pa = 0.5f * (lo_pa + hi_pa);
        const float t_qa = 0.5f * (lo_qa + hi_qa);
        const float t_pb = 0.5f * (lo_pb + hi_pb);
        const float t_qb = 0.5f * (lo_qb + hi_qb);
        const float fl_pa = fmaxf(-2.0f * t_pa, 0.0f);
        const float fl_qa = fmaxf(-2.0f * t_qa, 0.0f);
        const float fl_pb = fmaxf(-2.0f * t_pb, 0.0f);
        const float fl_qb = fmaxf(-2.0f * t_qb, 0.0f);
        float f_pa = 0.0f, f_qa = 0.0f, f_pb = 0.0f, f_qb = 0.0f;

        for (int h0 = 0; h0 < HP; h0 += 16) {
            float pv = pP[h0 + l16];
            float a0 = half ? 0.0f : pv;
            v2f Ap; Ap[0] =  a0; Ap[1] = 0.0f;
            v2f Aq; Aq[0] = -a0; Aq[1] = 0.0f;
            v8f Ca, Cb;
            #pragma unroll
            for (int r = 0; r < 8; ++r) {
                const float* row = &sW[(h0 + r + rowoff) * WPAD];
                Ca[r] = row[l16];
                Cb[r] = row[l16 + 16];
            }
            v8f Dpa = __builtin_amdgcn_wmma_f32_16x16x4_f32(false, Ap, false, Bm, (short)0, Ca, false, false);
            v8f Dqa = __builtin_amdgcn_wmma_f32_16x16x4_f32(false, Aq, false, Bm, (short)0, Ca, false, false);
            v8f Dpb = __builtin_amdgcn_wmma_f32_16x16x4_f32(false, Ap, false, Bm, (short)0, Cb, false, false);
            v8f Dqb = __builtin_amdgcn_wmma_f32_16x16x4_f32(false, Aq, false, Bm, (short)0, Cb, false, false);
            #pragma unroll
            for (int r = 0; r < 8; ++r) {
                f_pa += fmaxf(__builtin_fabsf(Dpa[r]) - t_pa, fl_pa);
                f_qa += fmaxf(__builtin_fabsf(Dqa[r]) - t_qa, fl_qa);
                f_pb += fmaxf(__builtin_fabsf(Dpb[r]) - t_pb, fl_pb);
                f_qb += fmaxf(__builtin_fabsf(Dqb[r]) - t_qb, fl_qb);
            }
        }
        // merge the two interleaved h-halves (lanes L and L^16 own the same o)
        f_pa += __shfl_xor(f_pa, 16, 32);
        f_qa += __shfl_xor(f_qa, 16, 32);
        f_pb += __shfl_xor(f_pb, 16, 32);
        f_qb += __shfl_xor(f_qb, 16, 32);

        lo_pa = (f_pa > GAMMA) ? t_pa : lo_pa;  hi_pa = (f_pa > GAMMA) ? hi_pa : t_pa;
        lo_qa = (f_qa > GAMMA) ? t_qa : lo_qa;  hi_qa = (f_qa > GAMMA) ? hi_qa : t_qa;
        lo_pb = (f_pb > GAMMA) ? t_pb : lo_pb;  hi_pb = (f_pb > GAMMA) ? hi_pb : t_pb;
        lo_qb = (f_qb > GAMMA) ? t_qb : lo_qb;  hi_qb = (f_qb > GAMMA) ? hi_qb : t_qb;
    }

    const float tau_pa = 0.5f * (lo_pa + hi_pa);
    const float tau_qa = 0.5f * (lo_qa + hi_qa);
    const float tau_pb = 0.5f * (lo_pb + hi_pb);
    const float tau_qb = 0.5f * (lo_qb + hi_qb);

    // lanes 0-15 own o = lane (half a); lanes 16-31 own o = lane (half b, o = 16+l16)
    const float tp = half ? tau_pb : tau_pa;
    const float tq = half ? tau_qb : tau_qa;
    const float conv = 0.5f * DTOT * (tp - tq);   // gamma*D*tau difference / 2

    out[(b * 32 + lane) * 1024 + pos] = conv;
}

extern "C" void kernel_launch(void* const* d_in, const int* in_sizes, int n_in,
                              void* d_out, int out_size, void* d_ws, size_t ws_size,
                              hipStream_t stream) {
    const float* x = (const float*)d_in[0];   // [2,32,32,32]
    const float* w = (const float*)d_in[1];   // [32,32,3,3]
    float* out = (float*)d_out;               // [2,32,32,32]
    (void)in_sizes; (void)n_in; (void)out_size; (void)d_ws; (void)ws_size;

    // 2048 locations, 8 per block (one wave32 each) -> 256 blocks of 256 threads
    mpconv_sparsemax_kernel<<<256, 256, 0, stream>>>(x, w, out);
}